// KernalAnsatz_65481071409588
// MI455X (gfx1250) — compile-verified
//
#include <hip/hip_runtime.h>
#include <math.h>

// ---------------------------------------------------------------------------
// KernalAnsatz: |<psi_x|psi_y>|^2 for psi = U(params) . (⊗_q Ry(a_q)) |0>.
//
// Key identity: U(params) is the SAME unitary for both states, so it cancels:
//     <psi_x|psi_y> = <0| Ry(x)^† Ry(y) |0> = prod_q cos((y_q - x_q)/2)
// => result = prod_q cos^2((y_q - x_q)/2).   (params provably unused)
//
// This replaces ~53 GB of statevector traffic (2 x 207 full-state gate passes
// over 64 MB states; >= 2.3 ms at 23.3 TB/s HBM) with 23 scalar cosines.
//
// The 23-way product is evaluated as exp(sum of logs); the cross-lane sum is
// done on the CDNA5 matrix unit with V_WMMA_F32_16X16X4_F32:
//   A[16x4]  : A[m,k] = 2*log|cos(d_{16k+m})|  (zero-padded past q=22)
//   B[4x16]  : all ones  ->  D[m,n] = v_m + v_{m+16} for every n
// Per the CDNA5 D-layout (8 VGPRs), lane 0 holds D[0..7][0] and lane 16 holds
// D[8..15][0]; one LDS combine + expf yields the answer.
// ---------------------------------------------------------------------------

typedef __attribute__((ext_vector_type(2))) float v2f;
typedef __attribute__((ext_vector_type(8))) float v8f;

#define N_QUBITS 23

__global__ __launch_bounds__(32)
void KernalAnsatz_overlap_kernel(const float* __restrict__ x,
                                 const float* __restrict__ y,
                                 float* __restrict__ out) {
    const int lane = threadIdx.x;  // wave32: exactly one wave in this block

    // ---- Build the A fragment of V_WMMA_F32_16X16X4_F32 ------------------
    // A (16x4 f32) VGPR layout (wave32):
    //   lanes 0-15 : a[0] = A[m=lane, k=0], a[1] = A[m=lane, k=1]
    //   lanes 16-31: a[0] = A[m=lane-16, k=2], a[1] = A[m=lane-16, k=3]
    // We map v_q -> A[m=q%16, k=q/16] (k in {0,1}), so lane L<16 computes the
    // two values it owns locally (q=L and q=L+16); lanes 16-31 carry zeros.
    v2f a; a.x = 0.0f; a.y = 0.0f;
    if (lane < 16) {
        {
            const int q = lane;                       // q in [0,16)
            const float d = 0.5f * (y[q] - x[q]);
            a.x = 2.0f * logf(fabsf(cosf(d)));        // log(cos^2); log(0)=-inf OK
        }
        const int q1 = lane + 16;                     // q in [16,32)
        if (q1 < N_QUBITS) {
            const float d = 0.5f * (y[q1] - x[q1]);
            a.y = 2.0f * logf(fabsf(cosf(d)));
        }
    }

    // B (4x16 f32) = all ones in every slot regardless of exact striping.
    v2f b; b.x = 1.0f; b.y = 1.0f;

    // D = A x B + 0 : D[m][n] = v_m + v_{m+16}  (independent of n).
    v8f c = {};
    // 8-arg form: (neg_a, A, neg_b, B, c_mod, C, reuse_a, reuse_b)
    c = __builtin_amdgcn_wmma_f32_16x16x4_f32(
            /*neg_a=*/false, a, /*neg_b=*/false, b,
            /*c_mod=*/(short)0, c, /*reuse_a=*/false, /*reuse_b=*/false);

    // Per-lane partial: lane in [0,16) sums rows 0..7 (of its column),
    // lane in [16,32) sums rows 8..15. Any single column n suffices.
    float s = c[0] + c[1] + c[2] + c[3] + c[4] + c[5] + c[6] + c[7];

    __shared__ float sm[32];
    sm[lane] = s;
    __syncthreads();

    if (lane == 0) {
        const float total = sm[0] + sm[16];  // = sum_q 2*log|cos(d_q)|
        out[0] = expf(total);                // = prod_q cos^2(d_q); exp(-inf)=0
    }
}

extern "C" void kernel_launch(void* const* d_in, const int* in_sizes, int n_in,
                              void* d_out, int out_size, void* d_ws, size_t ws_size,
                              hipStream_t stream) {
    const float* x = (const float*)d_in[0];   // (23,) float32
    const float* y = (const float*)d_in[1];   // (23,) float32
    // d_in[2] = params (2,23,3) float32 — provably unused: U(params)†U(params)=I
    float* out = (float*)d_out;               // scalar float32

    KernalAnsatz_overlap_kernel<<<dim3(1), dim3(32), 0, stream>>>(x, y, out);
}